// HybridGCNGATModel_4973572128957
// MI455X (gfx1250) — compile-verified
//
#include <hip/hip_runtime.h>
#include <hip/hip_bf16.h>

#define HID 64
#define HEADS 8
#define CH 8

typedef float v2f __attribute__((ext_vector_type(2)));
typedef float v8f __attribute__((ext_vector_type(8)));

// ---------------- small utility kernels ----------------

__global__ void fill_kernel(float* __restrict__ p, float v, int n) {
    int i = blockIdx.x * blockDim.x + threadIdx.x;
    if (i < n) p[i] = v;
}

__global__ void bias_init_kernel(float* __restrict__ out, const float* __restrict__ b, int n64) {
    int i = blockIdx.x * blockDim.x + threadIdx.x;
    if (i < n64) out[i] = b[i & (HID - 1)];
}

__global__ void relu_kernel(float* __restrict__ p, int n) {
    int i = blockIdx.x * blockDim.x + threadIdx.x;
    if (i < n) { float v = p[i]; p[i] = v > 0.f ? v : 0.f; }
}

// ---------------- degree / norm ----------------

__global__ void deg_kernel(const int* __restrict__ dst, float* __restrict__ deg, int E, int n) {
    int i = blockIdx.x * blockDim.x + threadIdx.x;
    int tot = E + n;
    if (i >= tot) return;
    int d = (i < E) ? dst[i] : (i - E);   // self-loops appended
    atomicAdd(&deg[d], 1.0f);
}

__global__ void dinv_kernel(float* __restrict__ deg, int n) {
    int i = blockIdx.x * blockDim.x + threadIdx.x;
    if (i >= n) return;
    float d = deg[i];
    deg[i] = d > 0.f ? rsqrtf(d) : 0.f;
}

// ---------------- WMMA fp32 GEMM: out[M,64] = A[M,K] @ W[K,64] ----------------
// One wave computes a full 16x64 output strip (4 16x16 tiles) with
// V_WMMA_F32_16X16X4_F32: A strip is loaded once per K-step and shared by
// the 4 WMMAs. 4 waves per block, each owning a different row strip.

__global__ void gemm_wmma_kernel(const float* __restrict__ A, const float* __restrict__ W,
                                 float* __restrict__ out, int M, int K) {
    const int lane = threadIdx.x & 31;   // 0..31
    const int wave = threadIdx.x >> 5;   // 0..3
    const int half = lane >> 4;          // 0: lanes 0-15, 1: lanes 16-31
    const int l16  = lane & 15;
    const int strip = blockIdx.x * 4 + wave;       // 16-row strip index
    int rowA = strip * 16 + l16;
    if (rowA >= M) rowA = M - 1;         // clamp: EXEC must stay all-ones for WMMA

    const float* __restrict__ Arow = A + (size_t)rowA * K;
    v8f c0 = {}, c1 = {}, c2 = {}, c3 = {};
    for (int k = 0; k < K; k += 4) {
        const int ka = k + half * 2;     // lanes 16-31 own K+2,K+3
        v2f a;
        a.x = Arow[ka];
        a.y = Arow[ka + 1];
        const float* __restrict__ Wr0 = W + (size_t)ka * HID;
        const float* __restrict__ Wr1 = Wr0 + HID;
        v2f b0, b1, b2, b3;
        b0.x = Wr0[l16];      b0.y = Wr1[l16];
        b1.x = Wr0[l16 + 16]; b1.y = Wr1[l16 + 16];
        b2.x = Wr0[l16 + 32]; b2.y = Wr1[l16 + 32];
        b3.x = Wr0[l16 + 48]; b3.y = Wr1[l16 + 48];
        c0 = __builtin_amdgcn_wmma_f32_16x16x4_f32(false, a, false, b0, (short)0, c0, false, false);
        c1 = __builtin_amdgcn_wmma_f32_16x16x4_f32(false, a, false, b1, (short)0, c1, false, false);
        c2 = __builtin_amdgcn_wmma_f32_16x16x4_f32(false, a, false, b2, (short)0, c2, false, false);
        c3 = __builtin_amdgcn_wmma_f32_16x16x4_f32(false, a, false, b3, (short)0, c3, false, false);
    }
    // C/D layout: VGPR r -> row strip*16 + r + half*8, col tileN + l16
    const int rowBase = strip * 16 + half * 8;
#pragma unroll
    for (int r = 0; r < 8; ++r) {
        const int row = rowBase + r;
        if (row < M) {
            float* o = out + (size_t)row * HID + l16;
            o[0]  = c0[r];
            o[16] = c1[r];
            o[32] = c2[r];
            o[48] = c3[r];
        }
    }
}

// ---------------- GCN scatter: out[dst] += h[src] * dinv[src]*dinv[dst] ----------------
// One wave per edge; each lane owns 2 channels (32*2 = 64).

__global__ void gcn_scatter_kernel(const float* __restrict__ h, const float* __restrict__ dinv,
                                   const int* __restrict__ src, const int* __restrict__ dst,
                                   float* __restrict__ out, int E, int n) {
    long long t = (long long)blockIdx.x * blockDim.x + threadIdx.x;
    int edge = (int)(t >> 5);
    int lane = (int)(t & 31);
    int tot = E + n;
    if (edge >= tot) return;
    int s, d;
    if (edge < E) { s = src[edge]; d = dst[edge]; }
    else          { s = edge - E;  d = s; }
    float norm = dinv[s] * dinv[d];
    const float2 hv = *(const float2*)(h + (size_t)s * HID + lane * 2);
    float* o = out + (size_t)d * HID + lane * 2;
    atomicAdd(o,     hv.x * norm);
    atomicAdd(o + 1, hv.y * norm);
}

// ---------------- GAT ----------------

__device__ __forceinline__ float leaky02(float x) { return x > 0.f ? x : 0.2f * x; }

__device__ __forceinline__ void atomicMaxF(float* addr, float v) {
    // IEEE-754 ordered-as-int trick
    if (v >= 0.f) atomicMax((int*)addr, __float_as_int(v));
    else          atomicMin((unsigned int*)addr, __float_as_uint(v));
}

__global__ void gat_alpha_kernel(const float* __restrict__ hg,
                                 const float* __restrict__ att_s,
                                 const float* __restrict__ att_d,
                                 float* __restrict__ asrc, float* __restrict__ adst, int n) {
    int i = blockIdx.x * blockDim.x + threadIdx.x;   // n*HEADS threads
    if (i >= n * HEADS) return;
    int node = i >> 3, head = i & 7;
    const float* hp = hg + (size_t)node * HID + head * CH;
    float ss = 0.f, sd = 0.f;
#pragma unroll
    for (int c = 0; c < CH; ++c) {
        float v = hp[c];
        ss += v * att_s[head * CH + c];
        sd += v * att_d[head * CH + c];
    }
    asrc[i] = ss; adst[i] = sd;
}

__global__ void gat_max_kernel(const float* __restrict__ asrc, const float* __restrict__ adst,
                               const int* __restrict__ src, const int* __restrict__ dst,
                               float* __restrict__ m, int E, int n) {
    long long t = (long long)blockIdx.x * blockDim.x + threadIdx.x;
    int edge = (int)(t >> 3), head = (int)(t & 7);
    int tot = E + n;
    if (edge >= tot) return;
    int s, d;
    if (edge < E) { s = src[edge]; d = dst[edge]; } else { s = edge - E; d = s; }
    float e = leaky02(asrc[s * 8 + head] + adst[d * 8 + head]);
    atomicMaxF(&m[d * 8 + head], e);
}

__global__ void gat_sum_kernel(const float* __restrict__ asrc, const float* __restrict__ adst,
                               const int* __restrict__ src, const int* __restrict__ dst,
                               const float* __restrict__ m, float* __restrict__ z, int E, int n) {
    long long t = (long long)blockIdx.x * blockDim.x + threadIdx.x;
    int edge = (int)(t >> 3), head = (int)(t & 7);
    int tot = E + n;
    if (edge >= tot) return;
    int s, d;
    if (edge < E) { s = src[edge]; d = dst[edge]; } else { s = edge - E; d = s; }
    float e = leaky02(asrc[s * 8 + head] + adst[d * 8 + head]);
    float p = __expf(e - m[d * 8 + head]);
    atomicAdd(&z[d * 8 + head], p);
}

__global__ void gat_scatter_kernel(const float* __restrict__ hg,
                                   const float* __restrict__ asrc, const float* __restrict__ adst,
                                   const float* __restrict__ m, const float* __restrict__ z,
                                   const int* __restrict__ src, const int* __restrict__ dst,
                                   float* __restrict__ out, int E, int n) {
    long long t = (long long)blockIdx.x * blockDim.x + threadIdx.x;
    int edge = (int)(t >> 5);
    int lane = (int)(t & 31);
    int tot = E + n;
    if (edge >= tot) return;
    int s, d;
    if (edge < E) { s = src[edge]; d = dst[edge]; } else { s = edge - E; d = s; }
    int head = lane >> 2;  // channels 2*lane, 2*lane+1 -> head = lane/4
    float e = leaky02(asrc[s * 8 + head] + adst[d * 8 + head]);
    float alpha = __expf(e - m[d * 8 + head]) / z[d * 8 + head];
    const float2 hv = *(const float2*)(hg + (size_t)s * HID + lane * 2);
    float* o = out + (size_t)d * HID + lane * 2;
    atomicAdd(o,     alpha * hv.x);
    atomicAdd(o + 1, alpha * hv.y);
}

// ---------------- orchestration ----------------

static inline unsigned gsz(long long t) { return (unsigned)((t + 255) / 256); }

extern "C" void kernel_launch(void* const* d_in, const int* in_sizes, int n_in,
                              void* d_out, int out_size, void* d_ws, size_t ws_size,
                              hipStream_t stream) {
    const float* x       = (const float*)d_in[0];
    const int*   ei      = (const int*)  d_in[1];
    const float* W0      = (const float*)d_in[2];
    const float* b0      = (const float*)d_in[3];
    const float* Wg      = (const float*)d_in[4];
    const float* att_src = (const float*)d_in[5];
    const float* att_dst = (const float*)d_in[6];
    const float* bg      = (const float*)d_in[7];
    const float* W2      = (const float*)d_in[8];
    const float* b2      = (const float*)d_in[9];
    const float* W3      = (const float*)d_in[10];
    const float* b3      = (const float*)d_in[11];

    const int E  = in_sizes[1] / 2;
    const int IN = 128;
    const int n  = in_sizes[0] / IN;
    const int Etot = E + n;
    const int* src = ei;
    const int* dst = ei + E;

    float* ws   = (float*)d_ws;
    float* dinv = ws;                              // n
    float* buf0 = dinv + n;                        // n*64 (GEMM outputs)
    float* buf1 = buf0 + (size_t)n * HID;          // n*64 (aggregate outputs)
    float* asrc = buf1 + (size_t)n * HID;          // n*8
    float* adst = asrc + (size_t)n * HEADS;        // n*8
    float* mbuf = adst + (size_t)n * HEADS;        // n*8
    float* zbuf = mbuf + (size_t)n * HEADS;        // n*8
    float* outp = (float*)d_out;

    const dim3 blk(256);
    const dim3 gblk(128);                               // 4 waves per block
    const int strips = (n + 15) / 16;
    const dim3 ggrid((strips + 3) / 4);

    // degrees -> dinv (in place)
    fill_kernel<<<gsz(n), blk, 0, stream>>>(dinv, 0.f, n);
    deg_kernel<<<gsz(Etot), blk, 0, stream>>>(dst, dinv, E, n);
    dinv_kernel<<<gsz(n), blk, 0, stream>>>(dinv, n);

    // ---- layer 0: GCN(128 -> 64) + relu ----
    gemm_wmma_kernel<<<ggrid, gblk, 0, stream>>>(x, W0, buf0, n, IN);
    bias_init_kernel<<<gsz((long long)n * HID), blk, 0, stream>>>(buf1, b0, n * HID);
    gcn_scatter_kernel<<<gsz((long long)Etot * 32), blk, 0, stream>>>(buf0, dinv, src, dst, buf1, E, n);
    relu_kernel<<<gsz((long long)n * HID), blk, 0, stream>>>(buf1, n * HID);

    // ---- layer 1: GAT(64 -> 8x8 concat) + relu ----
    gemm_wmma_kernel<<<ggrid, gblk, 0, stream>>>(buf1, Wg, buf0, n, HID);   // hg (no bias)
    gat_alpha_kernel<<<gsz((long long)n * HEADS), blk, 0, stream>>>(buf0, att_src, att_dst, asrc, adst, n);
    fill_kernel<<<gsz((long long)n * HEADS), blk, 0, stream>>>(mbuf, -1e30f, n * HEADS);
    fill_kernel<<<gsz((long long)n * HEADS), blk, 0, stream>>>(zbuf, 0.f, n * HEADS);
    gat_max_kernel<<<gsz((long long)Etot * 8), blk, 0, stream>>>(asrc, adst, src, dst, mbuf, E, n);
    gat_sum_kernel<<<gsz((long long)Etot * 8), blk, 0, stream>>>(asrc, adst, src, dst, mbuf, zbuf, E, n);
    bias_init_kernel<<<gsz((long long)n * HID), blk, 0, stream>>>(buf1, bg, n * HID);
    gat_scatter_kernel<<<gsz((long long)Etot * 32), blk, 0, stream>>>(buf0, asrc, adst, mbuf, zbuf,
                                                                      src, dst, buf1, E, n);
    relu_kernel<<<gsz((long long)n * HID), blk, 0, stream>>>(buf1, n * HID);

    // ---- layer 2: GCN(64 -> 64) + relu ----
    gemm_wmma_kernel<<<ggrid, gblk, 0, stream>>>(buf1, W2, buf0, n, HID);
    bias_init_kernel<<<gsz((long long)n * HID), blk, 0, stream>>>(buf1, b2, n * HID);
    gcn_scatter_kernel<<<gsz((long long)Etot * 32), blk, 0, stream>>>(buf0, dinv, src, dst, buf1, E, n);
    relu_kernel<<<gsz((long long)n * HID), blk, 0, stream>>>(buf1, n * HID);

    // ---- layer 3: GCN(64 -> 64), no relu, into d_out ----
    gemm_wmma_kernel<<<ggrid, gblk, 0, stream>>>(buf1, W3, buf0, n, HID);
    bias_init_kernel<<<gsz((long long)n * HID), blk, 0, stream>>>(outp, b3, n * HID);
    gcn_scatter_kernel<<<gsz((long long)Etot * 32), blk, 0, stream>>>(buf0, dinv, src, dst, outp, E, n);
}